// DecoderRNN_34059090657881
// MI455X (gfx1250) — compile-verified
//
#include <hip/hip_runtime.h>
#include <hip/hip_bf16.h>
#include <math.h>

// ---------------------------------------------------------------------------
// DecoderRNN (attention LSTM decoder, greedy decode) for MI455X / gfx1250.
// f32 matrix math via V_WMMA_F32_16X16X4_F32 (B=16 == WMMA M tile).
// Vocab projection streams out_w through LDS via async global->LDS copies
// (ASYNCcnt) with bank-conflict-free padded layouts.
// ---------------------------------------------------------------------------

typedef __attribute__((ext_vector_type(2))) float v2f;
typedef __attribute__((ext_vector_type(8))) float v8f;

#define VOCAB 32000
#define EMB   512
#define HID   512
#define BATCH 16
#define TSTEP 128

// ----- CDNA5 async global->LDS support (guarded; safe fallback) -------------
#if defined(__AMDGCN__)
typedef int v4i_vs __attribute__((vector_size(16)));
typedef __attribute__((address_space(1))) v4i_vs as1_v4i;
typedef __attribute__((address_space(3))) v4i_vs as3_v4i;
# if defined(__has_builtin)
#  if __has_builtin(__builtin_amdgcn_global_load_async_to_lds_b128)
#   define HAVE_ASYNC_LDS 1
#  endif
#  if __has_builtin(__builtin_amdgcn_s_wait_asynccnt)
#   define WAIT_ASYNCCNT(n) __builtin_amdgcn_s_wait_asynccnt(n)
#  endif
# endif
#endif
#ifndef HAVE_ASYNC_LDS
# define HAVE_ASYNC_LDS 0
#endif
#ifndef WAIT_ASYNCCNT
# if defined(__AMDGCN__) && HAVE_ASYNC_LDS
#  define WAIT_ASYNCCNT(n) asm volatile("s_wait_asynccnt %0" :: "i"(n) : "memory")
# else
#  define WAIT_ASYNCCNT(n)
# endif
#endif
#if HAVE_ASYNC_LDS
# define CHUNK_WAIT(n) WAIT_ASYNCCNT(n)
#else
# define CHUNK_WAIT(n)
#endif

// ---------------------------------------------------------------------------
// Init: copy initial h/c into scratch, set start tokens to 1.
// ---------------------------------------------------------------------------
__global__ void init_kernel(const float* __restrict__ h_in,
                            const float* __restrict__ c_in,
                            float* __restrict__ hc, int* __restrict__ tok) {
  int idx = blockIdx.x * blockDim.x + threadIdx.x;
  if (idx < 2 * BATCH * HID) {
    hc[idx]         = h_in[idx];   // h0 then h1 (matches [L,B,H] flat)
    hc[16384 + idx] = c_in[idx];   // c0 then c1
  }
  if (idx < BATCH) tok[idx] = 1;
}

// ---------------------------------------------------------------------------
// Attention (dot, softmax over S=128) + context + embedding lookup.
// ---------------------------------------------------------------------------
__global__ __launch_bounds__(128)
void attn_embed_kernel(const float* __restrict__ emb_table,
                       const float* __restrict__ enc,     // [S=128, B=16, H=512]
                       const float* __restrict__ h1,      // [16, 512]
                       const int*   __restrict__ tok,
                       float* __restrict__ x,             // [16, 1024]
                       float* __restrict__ attn_out,      // [B, T, S]
                       int t) {
  const int b = blockIdx.x;
  const int s = threadIdx.x;
  __shared__ float aw[128];
  __shared__ float red[128];

  const float* hq = h1 + b * HID;
  const float* es = enc + ((size_t)s * BATCH + b) * HID;
  float acc = 0.f;
  #pragma unroll 4
  for (int h = 0; h < HID; ++h) acc = fmaf(hq[h], es[h], acc);

  red[s] = acc;
  __syncthreads();
  for (int off = 64; off > 0; off >>= 1) {
    if (s < off) red[s] = fmaxf(red[s], red[s + off]);
    __syncthreads();
  }
  float m = red[0];
  __syncthreads();
  float ex = __expf(acc - m);
  red[s] = ex;
  __syncthreads();
  for (int off = 64; off > 0; off >>= 1) {
    if (s < off) red[s] += red[s + off];
    __syncthreads();
  }
  float w = ex / red[0];
  aw[s] = w;
  attn_out[((size_t)b * TSTEP + t) * 128 + s] = w;
  __syncthreads();

  const int tb = tok[b];
  for (int h = s; h < HID; h += 128) {
    float c = 0.f;
    #pragma unroll 4
    for (int ss = 0; ss < 128; ++ss)
      c = fmaf(aw[ss], enc[((size_t)ss * BATCH + b) * HID + h], c);
    x[b * (EMB + HID) + EMB + h] = c;
    x[b * (EMB + HID) + h] = emb_table[(size_t)tb * EMB + h];
  }
}

// ---------------------------------------------------------------------------
// Gates GEMM: out[16 x N] = A1[16 x K1] @ W1^T + A2[16 x K2] @ W2^T.
// A staged in LDS with +4 row padding (conflict-free ds_load_b64).
// ---------------------------------------------------------------------------
__global__ __launch_bounds__(256)
void gates_gemm_kernel(const float* __restrict__ A1, int K1,
                       const float* __restrict__ W1,
                       const float* __restrict__ A2, int K2,
                       const float* __restrict__ W2,
                       float* __restrict__ out, int N) {
  __shared__ float lds[16 * (1028 + 516)];   // K1<=1024, K2<=512, +4 pad each
  const int tid = threadIdx.x;
  const int K1S = K1 + 4, K2S = K2 + 4;

  for (int j = tid; j < 16 * K1 / 4; j += 256) {
    int row = j / (K1 / 4), col = (j % (K1 / 4)) * 4;
    *(float4*)(lds + row * K1S + col) = ((const float4*)A1)[j];
  }
  float* ldsB = lds + 16 * K1S;
  for (int j = tid; j < 16 * K2 / 4; j += 256) {
    int row = j / (K2 / 4), col = (j % (K2 / 4)) * 4;
    *(float4*)(ldsB + row * K2S + col) = ((const float4*)A2)[j];
  }
  __syncthreads();

  const int wave = tid >> 5, lane = tid & 31;
  const int n0   = (blockIdx.x * 8 + wave) * 16;
  const int mrow = lane & 15;
  const int kof  = (lane < 16) ? 0 : 2;       // lane-half selects K pair
  const int ncol = n0 + mrow;

  v8f c = {};
  const float* a1 = lds + mrow * K1S + kof;
  const float* w1 = W1 + (size_t)ncol * K1 + kof;
  for (int k = 0; k < K1; k += 4) {
    v2f a = *(const v2f*)(a1 + k);
    v2f b = *(const v2f*)(w1 + k);
    c = __builtin_amdgcn_wmma_f32_16x16x4_f32(false, a, false, b, (short)0, c,
                                              false, false);
  }
  const float* a2 = ldsB + mrow * K2S + kof;
  const float* w2 = W2 + (size_t)ncol * K2 + kof;
  for (int k = 0; k < K2; k += 4) {
    v2f a = *(const v2f*)(a2 + k);
    v2f b = *(const v2f*)(w2 + k);
    c = __builtin_amdgcn_wmma_f32_16x16x4_f32(false, a, false, b, (short)0, c,
                                              false, false);
  }

  const int mbase = (lane < 16) ? 0 : 8;
  #pragma unroll
  for (int r = 0; r < 8; ++r)
    out[(size_t)(mbase + r) * N + ncol] = c[r];
}

// ---------------------------------------------------------------------------
// LSTM cell elementwise (gate order i,f,g,o). 16*512 threads.
// ---------------------------------------------------------------------------
__global__ __launch_bounds__(256)
void lstm_cell_kernel(const float* __restrict__ gates,  // [16, 2048]
                      const float* __restrict__ bi,
                      const float* __restrict__ bh,
                      float* __restrict__ h, float* __restrict__ cst) {
  const int idx = blockIdx.x * blockDim.x + threadIdx.x;  // 0..8191
  const int b = idx >> 9, j = idx & 511;
  const float* g = gates + (size_t)b * 2048;
  float gi = g[j]        + bi[j]        + bh[j];
  float gf = g[512 + j]  + bi[512 + j]  + bh[512 + j];
  float gg = g[1024 + j] + bi[1024 + j] + bh[1024 + j];
  float go = g[1536 + j] + bi[1536 + j] + bh[1536 + j];
  float si = 1.f / (1.f + __expf(-gi));
  float sf = 1.f / (1.f + __expf(-gf));
  float so = 1.f / (1.f + __expf(-go));
  float cn = sf * cst[idx] + si * tanhf(gg);
  cst[idx] = cn;
  h[idx]   = so * tanhf(cn);
}

// ---------------------------------------------------------------------------
// Vocab projection: logits[16 x 32000] = h1[16x512] @ out_w^T + out_b.
// 250 blocks x 8 waves; each wave one 16-wide N tile.
// out_w streamed per wave through double-buffered LDS chunks (16 rows x 64 K)
// via async global->LDS b128 copies (coalesced, 4 lines/req), padded stride 68
// for conflict-free ds_load_b64.  A (h1) staged once per block, stride 516.
// ---------------------------------------------------------------------------
#define PCH   64                  // K-chunk
#define BSTR  (PCH + 4)           // 68: 68 mod 64 = 4 -> conflict-free
#define ASTR  (HID + 4)           // 516
#define BUF_F (16 * BSTR)         // 1088 floats per buffer

__device__ __forceinline__
void stage_chunk(const float* __restrict__ gsrc,  // outw + n0*HID + k0
                 float* __restrict__ ldst, int lane) {
  #pragma unroll
  for (int j = 0; j < 8; ++j) {
    const int row = j * 2 + (lane >> 4);
    const int col = (lane & 15) * 4;
    const float* g = gsrc + (size_t)row * HID + col;
    float* l = ldst + row * BSTR + col;
#if HAVE_ASYNC_LDS
    __builtin_amdgcn_global_load_async_to_lds_b128((as1_v4i*)g, (as3_v4i*)l,
                                                   0, 0);
#else
    *(float4*)l = *(const float4*)g;
#endif
  }
}

__global__ __launch_bounds__(256)
void proj_kernel(const float* __restrict__ h1,
                 const float* __restrict__ outw,   // [V, H] row-major
                 const float* __restrict__ outb,
                 float* __restrict__ logits) {
  __shared__ float lds[16 * ASTR + 8 * 2 * BUF_F];  // ~103 KB (<=320KB/WG)
  const int tid = threadIdx.x;

  // stage A = h1 [16 x 512] with +4 row padding
  for (int j = tid; j < BATCH * HID / 4; j += 256) {
    int row = j >> 7, col = (j & 127) * 4;
    *(float4*)(lds + row * ASTR + col) = ((const float4*)h1)[j];
  }
  __syncthreads();

  const int wave = tid >> 5, lane = tid & 31;
  const int n0   = (blockIdx.x * 8 + wave) * 16;
  const int mrow = lane & 15;
  const int kof  = (lane < 16) ? 0 : 2;
  const int ncol = n0 + mrow;

  float* buf0 = lds + 16 * ASTR + wave * 2 * BUF_F;
  float* buf1 = buf0 + BUF_F;
  const float* wsrc = outw + (size_t)n0 * HID;

  v8f c = {};
  stage_chunk(wsrc, buf0, lane);                    // chunk 0 in flight
  const int nchunks = HID / PCH;                    // 8
  for (int ci = 0; ci < nchunks; ++ci) {
    float* cur = (ci & 1) ? buf1 : buf0;
    if (ci + 1 < nchunks) {
      float* nxt = (ci & 1) ? buf0 : buf1;
      stage_chunk(wsrc + (ci + 1) * PCH, nxt, lane);
      CHUNK_WAIT(8);                                // chunk ci resident
    } else {
      CHUNK_WAIT(0);
    }
    const float* ap = lds + mrow * ASTR + ci * PCH + kof;
    const float* bp = cur + mrow * BSTR + kof;
    #pragma unroll
    for (int kk = 0; kk < PCH; kk += 4) {
      v2f a = *(const v2f*)(ap + kk);               // ds_load_b64, no conflicts
      v2f b = *(const v2f*)(bp + kk);               // ds_load_b64, no conflicts
      c = __builtin_amdgcn_wmma_f32_16x16x4_f32(false, a, false, b, (short)0, c,
                                                false, false);
    }
  }

  const float bias = outb[ncol];
  const int mbase = (lane < 16) ? 0 : 8;
  #pragma unroll
  for (int r = 0; r < 8; ++r)
    logits[(size_t)(mbase + r) * VOCAB + ncol] = c[r] + bias;
}

// ---------------------------------------------------------------------------
// log_softmax over V + first-index argmax; writes logp slice, seq, next token.
// ---------------------------------------------------------------------------
__global__ __launch_bounds__(256)
void logsoftmax_argmax_kernel(const float* __restrict__ logits,
                              float* __restrict__ logp,   // [B, T, V]
                              float* __restrict__ seq_out,// [B, T]
                              int* __restrict__ tok, int t) {
  const int b = blockIdx.x, tid = threadIdx.x;
  const float* row = logits + (size_t)b * VOCAB;
  __shared__ float rmax[256];
  __shared__ int   ridx[256];
  __shared__ float rsum[256];

  float m = -INFINITY;
  int mi = 0x7fffffff;
  for (int v = tid; v < VOCAB; v += 256) {
    float xv = row[v];
    if (xv > m) { m = xv; mi = v; }
  }
  rmax[tid] = m; ridx[tid] = mi;
  __syncthreads();
  for (int off = 128; off > 0; off >>= 1) {
    if (tid < off) {
      float m2 = rmax[tid + off]; int i2 = ridx[tid + off];
      if (m2 > rmax[tid] || (m2 == rmax[tid] && i2 < ridx[tid])) {
        rmax[tid] = m2; ridx[tid] = i2;
      }
    }
    __syncthreads();
  }
  const float gm = rmax[0];
  const int   gi = ridx[0];
  __syncthreads();

  float s = 0.f;
  for (int v = tid; v < VOCAB; v += 256) s += __expf(row[v] - gm);
  rsum[tid] = s;
  __syncthreads();
  for (int off = 128; off > 0; off >>= 1) {
    if (tid < off) rsum[tid] += rsum[tid + off];
    __syncthreads();
  }
  const float lse = gm + __logf(rsum[0]);

  float* dst = logp + ((size_t)b * TSTEP + t) * VOCAB;
  for (int v = tid; v < VOCAB; v += 256) dst[v] = row[v] - lse;
  if (tid == 0) {
    seq_out[b * TSTEP + t] = (float)gi;
    tok[b] = gi;
  }
}

// ---------------------------------------------------------------------------
// Host side: enqueue 128 sequential decode steps on `stream`.
// ---------------------------------------------------------------------------
extern "C" void kernel_launch(void* const* d_in, const int* in_sizes, int n_in,
                              void* d_out, int out_size, void* d_ws, size_t ws_size,
                              hipStream_t stream) {
  const float* embedding = (const float*)d_in[0];
  const float* h_in      = (const float*)d_in[1];
  const float* c_in      = (const float*)d_in[2];
  const float* enc       = (const float*)d_in[3];
  const float* w_ih0     = (const float*)d_in[4];
  const float* w_hh0     = (const float*)d_in[5];
  const float* b_ih0     = (const float*)d_in[6];
  const float* b_hh0     = (const float*)d_in[7];
  const float* w_ih1     = (const float*)d_in[8];
  const float* w_hh1     = (const float*)d_in[9];
  const float* b_ih1     = (const float*)d_in[10];
  const float* b_hh1     = (const float*)d_in[11];
  const float* out_w     = (const float*)d_in[12];
  const float* out_b     = (const float*)d_in[13];

  float* out  = (float*)d_out;
  float* logp = out;                                   // 16*128*32000
  float* seq  = out + (size_t)BATCH * TSTEP * VOCAB;   // 2048
  float* attn = seq + BATCH * TSTEP;                   // 16*128*128

  float* ws    = (float*)d_ws;
  float* hc    = ws;             // h0,h1,c0,c1 : 4 * 8192
  float* h0    = hc;
  float* h1    = hc + 8192;
  float* c0    = hc + 16384;
  float* c1    = hc + 24576;
  float* x     = ws + 32768;     // 16*1024
  float* gates = ws + 49152;     // 16*2048
  float* logit = ws + 81920;     // 16*32000
  int*   tok   = (int*)(ws + 593920);

  init_kernel<<<(2 * BATCH * HID + 255) / 256, 256, 0, stream>>>(h_in, c_in, hc, tok);

  for (int t = 0; t < TSTEP; ++t) {
    attn_embed_kernel<<<BATCH, 128, 0, stream>>>(embedding, enc, h1, tok, x, attn, t);
    gates_gemm_kernel<<<16, 256, 0, stream>>>(x, 1024, w_ih0, h0, 512, w_hh0,
                                              gates, 2048);
    lstm_cell_kernel<<<32, 256, 0, stream>>>(gates, b_ih0, b_hh0, h0, c0);
    gates_gemm_kernel<<<16, 256, 0, stream>>>(h0, 512, w_ih1, h1, 512, w_hh1,
                                              gates, 2048);
    lstm_cell_kernel<<<32, 256, 0, stream>>>(gates, b_ih1, b_hh1, h1, c1);
    proj_kernel<<<VOCAB / 128, 256, 0, stream>>>(h1, out_w, out_b, logit);
    logsoftmax_argmax_kernel<<<BATCH, 256, 0, stream>>>(logit, logp, seq, tok, t);
  }
}